// ModulatedConv2d_68959994905286
// MI455X (gfx1250) — compile-verified
//
#include <hip/hip_runtime.h>

typedef __attribute__((ext_vector_type(2))) float v2f;
typedef __attribute__((ext_vector_type(4))) float v4f;
typedef __attribute__((ext_vector_type(8))) float v8f;

#define B_      8
#define CIN     512
#define COUT    512
#define SDIM    512
#define H_      64
#define W_      64
#define HW      4096

#define GAIN_FC 0.04419417382415922f   // 1/sqrt(512)
#define WGAIN   0.01473139127471974f   // 1/sqrt(512*9)

#define MT        128                  // co rows per block
#define AS_STRIDE 20                   // padded LDS row stride for A
#define BSP_ROW   136                  // floats per k-pair row: 66 pos x 2 + pad

// async copy: 16B/lane global -> LDS, tracked by ASYNCcnt (CDNA5)
#define ASYNC_A(ldsoff, gptr)                                               \
    asm volatile("global_load_async_to_lds_b128 %0, %1, off"                \
                 :: "v"(ldsoff), "v"(gptr) : "memory")
#define WAIT_ASYNC() asm volatile("s_wait_asynccnt 0x0" ::: "memory")

// ---------------------------------------------------------------------------
__global__ void k_style(const float* __restrict__ style,
                        const float* __restrict__ aw,
                        const float* __restrict__ ab,
                        float* __restrict__ sg) {
    int i = blockIdx.x * blockDim.x + threadIdx.x;   // 4096
    int b = i >> 9, c = i & 511;
    float acc = 0.f;
    const float* st = style + b * SDIM;
    const float* wr = aw + c * SDIM;
    for (int d = 0; d < SDIM; ++d) acc += st[d] * wr[d];
    sg[i] = (acc * GAIN_FC + ab[c]) * WGAIN;
}

// ---------------------------------------------------------------------------
__global__ void k_wsq(const float* __restrict__ w, float* __restrict__ wsq) {
    int i = blockIdx.x * blockDim.x + threadIdx.x;   // 262144
    const float* p = w + (size_t)i * 9;
    float s = 0.f;
#pragma unroll
    for (int t = 0; t < 9; ++t) s += p[t] * p[t];
    wsq[i] = s;
}

// ---------------------------------------------------------------------------
__global__ void k_dec(const float* __restrict__ sg,
                      const float* __restrict__ wsq,
                      float* __restrict__ dec) {
    int i = blockIdx.x * blockDim.x + threadIdx.x;   // 4096
    int b = i >> 9, co = i & 511;
    const float* sgb = sg + b * CIN;
    const float* wq  = wsq + co * CIN;
    float s = 0.f;
    for (int ci = 0; ci < CIN; ++ci) {
        float m = sgb[ci];
        s += m * m * wq[ci];
    }
    dec[i] = rsqrtf(s + 1e-8f);
}

// ---------------------------------------------------------------------------
__global__ void k_wpack(const float* __restrict__ w, float* __restrict__ wT) {
    int i = blockIdx.x * blockDim.x + threadIdx.x;   // 9*512*512
    int tap = i >> 18;
    int rem = i & (262144 - 1);
    int co = rem >> 9, ci = rem & 511;
    wT[i] = w[(co * CIN + ci) * 9 + tap];
}

// ---------------------------------------------------------------------------
// Implicit-GEMM conv, fp32 WMMA 16x16x4.
// A-staging: async global->LDS (double-buffered). B-staging: VGPR prefetch
// with modulation multiply, pair-interleaved LDS layout.
// Block: 256 threads = 8 waves, 4(M) x 2(N); wave tile 32x32 (4 accums).
// grid = (COUT/128, H_, B_)
// ---------------------------------------------------------------------------
__global__ __launch_bounds__(256)
void k_conv(const float* __restrict__ x,
            const float* __restrict__ wT,
            const float* __restrict__ sg,
            const float* __restrict__ dec,
            float* __restrict__ out) {
    __shared__ float As[2][MT * AS_STRIDE];  // double-buffered, stride 20
    __shared__ float BsP[8 * BSP_ROW];       // [kpair][pos 0..65][k&1]

    const int tid  = threadIdx.x;
    const int lane = tid & 31;
    const int wv   = tid >> 5;
    const int wm   = wv >> 1;                // 0..3 -> 32-row co slice
    const int wn   = wv & 1;                 // 0..1 -> 32-col n slice
    const int l16  = lane & 15;
    const int lh   = lane >> 4;

    const int co_base = blockIdx.x * MT;
    const int h = blockIdx.y;
    const int b = blockIdx.z;

    const float* xb  = x  + b * (CIN * HW);
    const float* sgb = sg + b * CIN;

    // A-staging coords: 512 float4s, 2 async b128 per thread
    const int am    = tid >> 2;              // 0..63 (and +64)
    const int ak4   = (tid & 3) << 2;
    const int aofs0 = am * AS_STRIDE + ak4;
    const int aofs1 = (am + 64) * AS_STRIDE + ak4;
    const float* pA = wT + (co_base + am) * CIN + ak4;

    const unsigned ldsA0 = (unsigned)(uintptr_t)&As[0][aofs0];
    const unsigned ldsA1 = (unsigned)(uintptr_t)&As[0][aofs1];
    const unsigned bufB  = (unsigned)(MT * AS_STRIDE * 4);

    // B-staging coords
    const int bCi = tid >> 4;                // k row 0..15
    const int bN4 = (tid & 15) << 2;         // n start
    const int kpb = (bCi >> 1) * BSP_ROW + (bCi & 1);

    // constant-zero halo (pos 0 and 65, both parities, 8 kpairs)
    if (tid < 32) {
        int kp = tid >> 2, r = tid & 3;
        BsP[kp * BSP_ROW + ((r & 1) ? 65 : 0) * 2 + (r >> 1)] = 0.f;
    }

    v8f acc00 = {0.f,0.f,0.f,0.f,0.f,0.f,0.f,0.f};
    v8f acc01 = acc00, acc10 = acc00, acc11 = acc00;

    const int dhlo = (h == 0) ? 0 : -1;
    const int dhhi = (h == H_ - 1) ? 0 : 1;
    int p = 0;                               // A buffer parity (uniform)

    for (int dh = dhlo; dh <= dhhi; ++dh) {
        const float* xrowp = xb + (h + dh) * W_ + bCi * HW + bN4;
        const int tapbase = (dh + 1) * 3;

        // prologue: async A for step (cc=0, dwi=0) into buffer p; B regs
        {
            const float* ga = pA + tapbase * (COUT * CIN);
            const unsigned lb = p * bufB;
            ASYNC_A(ldsA0 + lb, ga);
            ASYNC_A(ldsA1 + lb, ga + 64 * CIN);
        }
        v4f bv  = *(const v4f*)(xrowp);
        float sv = sgb[bCi];

        for (int cc = 0; cc < CIN; cc += 16) {
#pragma unroll
            for (int dwi = 0; dwi < 3; ++dwi) {
                __syncthreads();                   // prev compute done reading
                if (dwi == 0) {                    // publish modulated B row
                    v4f t = bv * sv;
                    BsP[kpb + (1 + bN4 + 0) * 2] = t.x;
                    BsP[kpb + (1 + bN4 + 1) * 2] = t.y;
                    BsP[kpb + (1 + bN4 + 2) * 2] = t.z;
                    BsP[kpb + (1 + bN4 + 3) * 2] = t.w;
                }
                WAIT_ASYNC();                      // this wave's A arrived
                __syncthreads();                   // A(buf p) + B visible

                const bool lastStep = (dwi == 2) && (cc + 16 == CIN);
                if (!lastStep) {                   // uniform
                    const int ntap = tapbase + ((dwi == 2) ? 0 : dwi + 1);
                    const int ncc  = (dwi == 2) ? cc + 16 : cc;
                    const float* ga = pA + ntap * (COUT * CIN) + ncc;
                    const unsigned lb = (p ^ 1) * bufB;
                    ASYNC_A(ldsA0 + lb, ga);
                    ASYNC_A(ldsA1 + lb, ga + 64 * CIN);
                    if (dwi == 2) {
                        bv = *(const v4f*)(xrowp + (cc + 16) * HW);
                        sv = sgb[cc + 16 + bCi];
                    }
                }

                const float* asA = &As[0][0] + p * (MT * AS_STRIDE)
                                 + (wm * 32 + l16) * AS_STRIDE;
                const float* bsB = &BsP[(dwi + wn * 32 + l16) * 2];
#pragma unroll
                for (int ks = 0; ks < 16; ks += 4) {
                    const int k0 = ks + lh * 2;    // always even
                    v2f af0 = *(const v2f*)(asA + k0);
                    v2f af1 = *(const v2f*)(asA + 16 * AS_STRIDE + k0);
                    v2f bf0 = *(const v2f*)(bsB + (k0 >> 1) * BSP_ROW);
                    v2f bf1 = *(const v2f*)(bsB + (k0 >> 1) * BSP_ROW + 32);

                    acc00 = __builtin_amdgcn_wmma_f32_16x16x4_f32(
                        false, af0, false, bf0, (short)0, acc00, false, false);
                    acc01 = __builtin_amdgcn_wmma_f32_16x16x4_f32(
                        false, af0, false, bf1, (short)0, acc01, false, false);
                    acc10 = __builtin_amdgcn_wmma_f32_16x16x4_f32(
                        false, af1, false, bf0, (short)0, acc10, false, false);
                    acc11 = __builtin_amdgcn_wmma_f32_16x16x4_f32(
                        false, af1, false, bf1, (short)0, acc11, false, false);
                }
                p ^= 1;
            }
        }
    }

    // ---- demodulate + store ----
    const float* decb = dec + b * COUT;
    float* ob = out + (b * COUT + co_base) * HW + h * W_;
    const int n0 = wn * 32 + l16;
#pragma unroll
    for (int r = 0; r < 8; ++r) {
        const int m0 = wm * 32 + r + lh * 8;
        const int m1 = m0 + 16;
        const float d0 = decb[co_base + m0];
        const float d1 = decb[co_base + m1];
        ob[m0 * HW + n0]      = acc00[r] * d0;
        ob[m0 * HW + n0 + 16] = acc01[r] * d0;
        ob[m1 * HW + n0]      = acc10[r] * d1;
        ob[m1 * HW + n0 + 16] = acc11[r] * d1;
    }
}

// ---------------------------------------------------------------------------
extern "C" void kernel_launch(void* const* d_in, const int* in_sizes, int n_in,
                              void* d_out, int out_size, void* d_ws, size_t ws_size,
                              hipStream_t stream) {
    const float* x   = (const float*)d_in[0];
    const float* sty = (const float*)d_in[1];
    const float* w   = (const float*)d_in[2];
    const float* aw  = (const float*)d_in[3];
    const float* ab  = (const float*)d_in[4];
    float* out = (float*)d_out;

    float* sg  = (float*)d_ws;                       // 4096 floats
    float* wsq = sg + B_ * CIN;                      // 262144 floats
    float* dec = wsq + COUT * CIN;                   // 4096 floats
    float* wT  = dec + B_ * COUT;                    // 9*512*512 floats (~9.4 MB)

    k_style<<<(B_ * CIN) / 256, 256, 0, stream>>>(sty, aw, ab, sg);
    k_wsq<<<(COUT * CIN) / 256, 256, 0, stream>>>(w, wsq);
    k_dec<<<(B_ * COUT) / 256, 256, 0, stream>>>(sg, wsq, dec);
    k_wpack<<<(9 * COUT * CIN) / 256, 256, 0, stream>>>(w, wT);

    dim3 grid(COUT / MT, H_, B_);
    k_conv<<<grid, 256, 0, stream>>>(x, wT, sg, dec, out);
}